// AttentionJointNet_3925600108642
// MI455X (gfx1250) — compile-verified
//
#include <hip/hip_runtime.h>
#include <hip/hip_bf16.h>
#include <math.h>

// Problem constants (fixed by the reference harness)
#define TT 1024
#define BB 4
#define EE 512
#define UU 64
#define DD 512
#define HH 1024
#define GG 128
#define NEGBIG (-1.0e30f)

typedef __attribute__((ext_vector_type(2))) float v2f;
typedef __attribute__((ext_vector_type(8))) float v8f;

__device__ __forceinline__ v8f wmma4(v2f a, v2f b, v8f c) {
  // D = A(16x4 f32) * B(4x16 f32) + C(16x16 f32)
  return __builtin_amdgcn_wmma_f32_16x16x4_f32(
      /*neg_a=*/false, a, /*neg_b=*/false, b,
      /*c_mod=*/(short)0, c, /*reuse_a=*/false, /*reuse_b=*/false);
}

// Generic C = A @ Bm^T + bias, all row-major. One wave computes a 16(M) x 64(N)
// tile. Fragment layout for V_WMMA_F32_16X16X4_F32: lanes 0-15 hold K=k0,k0+1,
// lanes 16-31 hold K=k0+2,k0+3 -> one b64 load per fragment, straight into the
// WMMA operand. Double-buffered: next K-step's 5 loads issue before the current
// step's 4 WMMAs (overlapped loadcnt waits); pointer-increment addressing so
// loads strength-reduce to immediate-offset form.
// amode==1: A rows follow the (T,B,E)->(B,T,E) transpose of encoder_state.
__global__ __launch_bounds__(256) void wgemm_nt(
    const float* __restrict__ A, int lda, int amode,
    const float* __restrict__ Bm, int ldb,
    const float* __restrict__ bias,
    float* __restrict__ C, int ldc,
    int M, int N, int K)
{
  const int lane = threadIdx.x & 31;
  const int half = lane >> 4;
  const int lr   = lane & 15;
  const int tho  = half << 1;          // lane-half K offset (0 or 2)
  const int wave = blockIdx.x * (blockDim.x >> 5) + (threadIdx.x >> 5);
  const int n64  = N >> 6;
  const int mt16 = M >> 4;
  if (wave >= mt16 * n64) return;
  const int mt = wave / n64;
  const int nt = wave % n64;

  const int row = mt * 16 + lr;
  long aoff;
  if (amode == 1)
    aoff = (long)(row & (TT - 1)) * (BB * EE) + (long)(row >> 10) * EE; // enc[t,b,:]
  else
    aoff = (long)row * lda;
  const float* Ar  = A + aoff + tho;
  const float* Br0 = Bm + (long)(nt * 64 +  0 + lr) * ldb + tho;
  const float* Br1 = Bm + (long)(nt * 64 + 16 + lr) * ldb + tho;
  const float* Br2 = Bm + (long)(nt * 64 + 32 + lr) * ldb + tho;
  const float* Br3 = Bm + (long)(nt * 64 + 48 + lr) * ldb + tho;

  v8f c0 = {}, c1 = {}, c2 = {}, c3 = {};

  v2f af = *(const v2f*)(Ar);
  v2f b0 = *(const v2f*)(Br0);
  v2f b1 = *(const v2f*)(Br1);
  v2f b2 = *(const v2f*)(Br2);
  v2f b3 = *(const v2f*)(Br3);

#pragma unroll 4
  for (int k0 = 4; k0 < K; k0 += 4) {
    const v2f afn = *(const v2f*)(Ar  + k0);
    const v2f b0n = *(const v2f*)(Br0 + k0);
    const v2f b1n = *(const v2f*)(Br1 + k0);
    const v2f b2n = *(const v2f*)(Br2 + k0);
    const v2f b3n = *(const v2f*)(Br3 + k0);
    c0 = wmma4(af, b0, c0);
    c1 = wmma4(af, b1, c1);
    c2 = wmma4(af, b2, c2);
    c3 = wmma4(af, b3, c3);
    af = afn; b0 = b0n; b1 = b1n; b2 = b2n; b3 = b3n;
  }
  c0 = wmma4(af, b0, c0);
  c1 = wmma4(af, b1, c1);
  c2 = wmma4(af, b2, c2);
  c3 = wmma4(af, b3, c3);

  const float bs0 = bias ? bias[nt * 64 +  0 + lr] : 0.0f;
  const float bs1 = bias ? bias[nt * 64 + 16 + lr] : 0.0f;
  const float bs2 = bias ? bias[nt * 64 + 32 + lr] : 0.0f;
  const float bs3 = bias ? bias[nt * 64 + 48 + lr] : 0.0f;
#pragma unroll
  for (int i = 0; i < 8; ++i) {
    const long m = mt * 16 + i + half * 8;      // C row held by this VGPR/lane-half
    float* Crow = C + m * (long)ldc + nt * 64 + lr;
    Crow[ 0] = c0[i] + bs0;
    Crow[16] = c1[i] + bs1;
    Crow[32] = c2[i] + bs2;
    Crow[48] = c3[i] + bs3;
  }
}

// Online prefix-softmax attention + tanh joint. One wave owns (b, 16 u, 64 h).
// Sequential sweep over groups g (8 new t per group), flash-attention style
// rescale, WMMA outer-product accumulation, per-group output slab.
// No local arrays (ext-vectors only); pointer-based addressing.
__global__ __launch_bounds__(256) void prefix_attn_out(
    const float* __restrict__ attn,          // (B,U,T)
    const float* __restrict__ Vp,            // (B,T,H)
    const float* __restrict__ hdec,          // (B,U,H)
    const unsigned char* __restrict__ mask,  // (B,T) bool
    float* __restrict__ out)                 // (B,G,U,H)
{
  const int lane = threadIdx.x & 31;
  const int half = lane >> 4;
  const int lr   = lane & 15;
  const int wave = blockIdx.x * (blockDim.x >> 5) + (threadIdx.x >> 5);
  if (wave >= BB * (UU / 16) * (HH / 64)) return;
  const int b  = wave >> 6;
  const int ut = (wave >> 4) & 3;
  const int ht = wave & 15;
  const int u0 = ut * 16;
  const int h0 = ht * 64;

  const float* arow = attn + ((long)b * UU + u0 + lr) * TT;   // this lane's u-row
  const float* vb   = Vp + (long)b * TT * HH + h0 + lr + (half << 1) * HH;
  const float* hb   = hdec + (long)b * UU * HH;
  const unsigned char* mb = mask + b * TT;
  float* ob = out + (long)b * GG * UU * HH + h0 + lr;

  v8f hd0, hd1, hd2, hd3;   // h_dec fragments (C-tile layout), register resident
#pragma unroll
  for (int i = 0; i < 8; ++i) {
    const long ur = (long)(u0 + i + half * 8) * HH + h0 + lr;
    hd0[i] = hb[ur];      hd1[i] = hb[ur + 16];
    hd2[i] = hb[ur + 32]; hd3[i] = hb[ur + 48];
  }

  float m = NEGBIG, l = 0.0f;
  v8f c0 = {}, c1 = {}, c2 = {}, c3 = {};

  for (int g = 0; g < GG; ++g) {
    const int t0 = g * 8;
    float4 av0 = *(const float4*)(arow + t0);
    float4 av1 = *(const float4*)(arow + t0 + 4);
    const uint2 mv = *(const uint2*)(mb + t0);
    if (g + 1 < GG) __builtin_prefetch(arow + t0 + 8, 0, 1);

    // encoder padding -> large negative
    if (mv.x & 0x000000ffu) av0.x = NEGBIG;
    if (mv.x & 0x0000ff00u) av0.y = NEGBIG;
    if (mv.x & 0x00ff0000u) av0.z = NEGBIG;
    if (mv.x & 0xff000000u) av0.w = NEGBIG;
    if (mv.y & 0x000000ffu) av1.x = NEGBIG;
    if (mv.y & 0x0000ff00u) av1.y = NEGBIG;
    if (mv.y & 0x00ff0000u) av1.z = NEGBIG;
    if (mv.y & 0xff000000u) av1.w = NEGBIG;

    const float cm = fmaxf(fmaxf(fmaxf(av0.x, av0.y), fmaxf(av0.z, av0.w)),
                           fmaxf(fmaxf(av1.x, av1.y), fmaxf(av1.z, av1.w)));
    const float mnew = fmaxf(m, cm);
    const float r = __expf(m - mnew);
    const float p0 = __expf(av0.x - mnew);
    const float p1 = __expf(av0.y - mnew);
    const float p2 = __expf(av0.z - mnew);
    const float p3 = __expf(av0.w - mnew);
    const float p4 = __expf(av1.x - mnew);
    const float p5 = __expf(av1.y - mnew);
    const float p6 = __expf(av1.z - mnew);
    const float p7 = __expf(av1.w - mnew);
    m = mnew;
    l = l * r + (((p0 + p1) + (p2 + p3)) + ((p4 + p5) + (p6 + p7)));

    // rescale accumulators: broadcast per-C-row factor (row = i + 8*half)
#pragma unroll
    for (int i = 0; i < 8; ++i) {
      const float ri = __shfl(r, (half << 3) + i, 32);
      c0[i] *= ri; c1[i] *= ri; c2[i] *= ri; c3[i] *= ri;
    }

    // A fragments: softmax weights for K=t0..t0+3 and t0+4..t0+7
    v2f af0; af0.x = half ? p2 : p0; af0.y = half ? p3 : p1;
    v2f af1; af1.x = half ? p6 : p4; af1.y = half ? p7 : p5;

    // B fragments: v rows t0+tho,t0+tho+1 (and +4) at 4 column tiles
    const float* vr = vb + (long)t0 * HH;
    v2f bf;
    bf.x = vr[         0]; bf.y = vr[1 * HH +  0];
    c0 = wmma4(af0, bf, c0);
    bf.x = vr[4 * HH + 0]; bf.y = vr[5 * HH +  0];
    c0 = wmma4(af1, bf, c0);
    bf.x = vr[        16]; bf.y = vr[1 * HH + 16];
    c1 = wmma4(af0, bf, c1);
    bf.x = vr[4 * HH + 16]; bf.y = vr[5 * HH + 16];
    c1 = wmma4(af1, bf, c1);
    bf.x = vr[        32]; bf.y = vr[1 * HH + 32];
    c2 = wmma4(af0, bf, c2);
    bf.x = vr[4 * HH + 32]; bf.y = vr[5 * HH + 32];
    c2 = wmma4(af1, bf, c2);
    bf.x = vr[        48]; bf.y = vr[1 * HH + 48];
    c3 = wmma4(af0, bf, c3);
    bf.x = vr[4 * HH + 48]; bf.y = vr[5 * HH + 48];
    c3 = wmma4(af1, bf, c3);

    // emit this group's 16x64 slab: tanh(acc/l + h_dec)
    const float il = 1.0f / l;
    float* og = ob + (long)g * UU * HH;
#pragma unroll
    for (int i = 0; i < 8; ++i) {
      const float li = __shfl(il, (half << 3) + i, 32);
      const long uo = (long)(u0 + i + half * 8) * HH;
      og[uo +  0] = tanhf(c0[i] * li + hd0[i]);
      og[uo + 16] = tanhf(c1[i] * li + hd1[i]);
      og[uo + 32] = tanhf(c2[i] * li + hd2[i]);
      og[uo + 48] = tanhf(c3[i] * li + hd3[i]);
    }
  }
}

__global__ void glen_kernel(const unsigned char* __restrict__ mask,
                            long long* __restrict__ out)
{
  const int b = threadIdx.x;
  if (b >= BB) return;
  int s = 0;
  for (int t = 0; t < TT; ++t) s += (mask[b * TT + t] ? 0 : 1);
  out[b] = (long long)((s + 7) / 8);   // ceil(len / DOWNSAMPLE)
}

extern "C" void kernel_launch(void* const* d_in, const int* in_sizes, int n_in,
                              void* d_out, int out_size, void* d_ws, size_t ws_size,
                              hipStream_t stream) {
  (void)in_sizes; (void)n_in; (void)out_size; (void)ws_size;
  const float* enc = (const float*)d_in[0];              // (T,B,E)
  const unsigned char* mask = (const unsigned char*)d_in[1]; // (B,T) bool
  const float* dec = (const float*)d_in[2];              // (B,U,D)
  const float* Wq = (const float*)d_in[3];  const float* bq = (const float*)d_in[4];
  const float* Wk = (const float*)d_in[5];  const float* bk = (const float*)d_in[6];
  const float* Wv = (const float*)d_in[7];  const float* bv = (const float*)d_in[8];
  const float* Wd = (const float*)d_in[9];  const float* bd = (const float*)d_in[10];
  float* out = (float*)d_out;

  // Workspace layout (floats): q | hdec | k | v | attn  (~35 MB total)
  float* ws   = (float*)d_ws;
  float* q    = ws;                    // 256*1024
  float* hd   = ws + 262144;           // 256*1024
  float* kbuf = ws + 524288;           // 4*1024*1024
  float* vbuf = ws + 4718592;          // 4*1024*1024
  float* attn = ws + 8912896;          // 4*64*1024

  auto nblocks = [](int waves) { return (waves + 7) / 8; };

  // Projections: q, h_dec (M=256,K=512,N=1024); k, v (M=4096,K=512,N=1024)
  wgemm_nt<<<nblocks(16 * 16),  256, 0, stream>>>(dec, DD, 0, Wq, DD, bq, q,    HH, BB * UU, HH, DD);
  wgemm_nt<<<nblocks(16 * 16),  256, 0, stream>>>(dec, DD, 0, Wd, DD, bd, hd,   HH, BB * UU, HH, DD);
  wgemm_nt<<<nblocks(256 * 16), 256, 0, stream>>>(enc, EE, 1, Wk, EE, bk, kbuf, HH, BB * TT, HH, EE);
  wgemm_nt<<<nblocks(256 * 16), 256, 0, stream>>>(enc, EE, 1, Wv, EE, bv, vbuf, HH, BB * TT, HH, EE);

  // attn[b] = q[b] @ k[b]^T  (M=64, N=1024, K=1024)
  for (int b = 0; b < BB; ++b) {
    wgemm_nt<<<nblocks(4 * 16), 256, 0, stream>>>(
        q + (long)b * UU * HH, HH, 0,
        kbuf + (long)b * TT * HH, HH,
        nullptr,
        attn + (long)b * UU * TT, TT,
        UU, TT, HH);
  }

  // Fused prefix-softmax PV + tanh joint  (256 waves = 32 blocks x 8 waves)
  prefix_attn_out<<<32, 256, 0, stream>>>(attn, vbuf, hd, mask, out);

  // group_lengths (int64 x B) appended after h_joint
  glen_kernel<<<1, 32, 0, stream>>>(mask, (long long*)(out + (long)BB * GG * UU * HH));
}